// Experts_39324720562908
// MI455X (gfx1250) — compile-verified
//
#include <hip/hip_runtime.h>
#include <hip/hip_bf16.h>

// MoE grouped FFN: out = gelu(x_e @ W1_e + b1_e) @ W2_e + b2_e, per expert.
// E=32, C=64 rows/expert, D=1024, F=4096, fp32.
// HBM-bound (~1.07 GB weights @ 23.3 TB/s => ~46us floor, AI ~32 FLOP/B):
// keep fp32 exactness, V_WMMA_F32_16X16X4_F32 for math, and stream weights
// with double-buffered GLOBAL_LOAD_ASYNC_TO_LDS_B128 (ASYNCcnt) fills.

#define EXPERTS 32
#define CAP     64
#define DMODEL  1024
#define DFF     4096

typedef __attribute__((ext_vector_type(2))) float v2f;
typedef __attribute__((ext_vector_type(8))) float v8f;

#define TILE_N   128            // output columns per block
#define KCHUNK   32             // K staged per LDS chunk (double buffered)
#define A_STR    36             // A tile stride (dwords): 4*9, b64-conflict-free, 16B aligned
#define B_STR    132            // B tile stride (dwords): 128+4, 16B aligned

template <int KTOT, int NTOT, bool GELU>
__global__ __launch_bounds__(256)
void expert_gemm(const float* __restrict__ A,     // [E, 64, KTOT]
                 const float* __restrict__ B,     // [E, KTOT, NTOT]
                 const float* __restrict__ bias,  // [E, NTOT]
                 float* __restrict__ Cout)        // [E, 64, NTOT]
{
    constexpr int NTILES = NTOT / TILE_N;
    constexpr int NCHUNK = KTOT / KCHUNK;
    const int e     = blockIdx.x / NTILES;
    const int nBase = (blockIdx.x % NTILES) * TILE_N;

    const int tid  = threadIdx.x;
    const int wave = tid >> 5;          // 8 waves
    const int lane = tid & 31;
    const int l15  = lane & 15;
    const int kh   = lane >> 4;         // K-half selector for 32-bit WMMA operands
    const int nOff = wave * 16;         // wave's 16-column strip of the 128 tile

    __shared__ alignas(16) float As[2][CAP * A_STR];     // [64 rows][36]
    __shared__ alignas(16) float Bs[2][KCHUNK * B_STR];  // [32 k][132] row-major

    const float* Ae = A + (size_t)e * CAP * KTOT;
    const float* Be = B + (size_t)e * KTOT * NTOT + nBase;
    const float* be = bias + (size_t)e * NTOT + nBase;
    float*       Ce = Cout + (size_t)e * CAP * NTOT + nBase;

    // Issue one chunk's worth of async global->LDS copies (no VGPR staging).
    // GVS mode: mem = SGPR64 base + per-lane u32 byte offset.
    auto fill_async = [&](int buf, int kc) {
        // A: 64 rows x 32 k = 512 float4 -> 2 per thread
#pragma unroll
        for (int i = 0; i < 2; ++i) {
            const int idx = tid + i * 256;
            const int row = idx >> 3;           // 8 float4 per row
            const int c4  = idx & 7;
            uint32_t lds = (uint32_t)(uintptr_t)&As[buf][row * A_STR + c4 * 4];
            uint32_t off = (uint32_t)((row * KTOT + kc + c4 * 4) * 4);
            asm volatile("global_load_async_to_lds_b128 %0, %1, %2"
                         :: "v"(lds), "v"(off), "s"(Ae) : "memory");
        }
        // B: 32 k x 128 n = 1024 float4 -> 4 per thread
#pragma unroll
        for (int i = 0; i < 4; ++i) {
            const int idx = tid + i * 256;
            const int k   = idx >> 5;           // 32 float4 per k-row
            const int n4  = idx & 31;
            uint32_t lds = (uint32_t)(uintptr_t)&Bs[buf][k * B_STR + n4 * 4];
            uint32_t off = (uint32_t)(((kc + k) * NTOT + n4 * 4) * 4);
            asm volatile("global_load_async_to_lds_b128 %0, %1, %2"
                         :: "v"(lds), "v"(off), "s"(Be) : "memory");
        }
    };

    v8f acc[4] = {};
    fill_async(0, 0);

    for (int c = 0; c < NCHUNK; ++c) {
        const int cb = c & 1;
        // Drain my async fills, then barrier: guarantees (a) current buffer
        // fully written by all waves, (b) all waves done computing on the
        // other buffer (previous iteration) -> safe to refill it.
        asm volatile("s_wait_asynccnt 0x0" ::: "memory");
        __syncthreads();
        if (c + 1 < NCHUNK) fill_async(cb ^ 1, (c + 1) * KCHUNK);

        const float* Ab = As[cb];
        const float* Bb = Bs[cb];
#pragma unroll
        for (int ks = 0; ks < KCHUNK; ks += 4) {
            // B 4x16 operand: lane -> col N=l15, K = 2*kh + {0,1}  (two rows,
            // stride B_STR apart -> ds_load_2addr_b32)
            v2f b;
            b.x = Bb[(ks + 2 * kh + 0) * B_STR + nOff + l15];
            b.y = Bb[(ks + 2 * kh + 1) * B_STR + nOff + l15];
#pragma unroll
            for (int m = 0; m < 4; ++m) {
                // A 16x4 operand: lane -> row M=l15, K = 2*kh + {0,1} (b64)
                v2f a = *(const v2f*)(&Ab[(m * 16 + l15) * A_STR + ks + 2 * kh]);
                acc[m] = __builtin_amdgcn_wmma_f32_16x16x4_f32(
                    false, a, false, b, (short)0, acc[m], false, false);
            }
        }
    }

    // Epilogue: bias (+ tanh-GELU), store.
    // C/D layout: VGPR r, lane l -> row M = r + 8*(l>>4), col N = l&15.
    const int   col = nOff + l15;
    const float bv  = be[col];
#pragma unroll
    for (int m = 0; m < 4; ++m) {
#pragma unroll
        for (int r = 0; r < 8; ++r) {
            const int row = m * 16 + r + 8 * kh;
            float v = acc[m][r] + bv;
            if (GELU) {
                const float u = 0.7978845608028654f * (v + 0.044715f * v * v * v);
                v = 0.5f * v * (1.0f + tanhf(u));
            }
            Ce[(size_t)row * NTOT + col] = v;
        }
    }
}

extern "C" void kernel_launch(void* const* d_in, const int* in_sizes, int n_in,
                              void* d_out, int out_size, void* d_ws, size_t ws_size,
                              hipStream_t stream) {
    const float* x  = (const float*)d_in[0];   // [1, E*C, D]
    const float* W1 = (const float*)d_in[1];   // [E, D, F]
    const float* b1 = (const float*)d_in[2];   // [E, F]
    const float* W2 = (const float*)d_in[3];   // [E, F, D]
    const float* b2 = (const float*)d_in[4];   // [E, D]
    float* out = (float*)d_out;                // [1, E*C, D]
    float* h   = (float*)d_ws;                 // [E, C, F] fp32 = 32 MB

    dim3 block(256);
    expert_gemm<DMODEL, DFF, true>
        <<<dim3(EXPERTS * (DFF / TILE_N)), block, 0, stream>>>(x, W1, b1, h);
    expert_gemm<DFF, DMODEL, false>
        <<<dim3(EXPERTS * (DMODEL / TILE_N)), block, 0, stream>>>(h, W2, b2, out);
}